// StructNConv2D_s_with_d_64544768524774
// MI455X (gfx1250) — compile-verified
//
#include <hip/hip_runtime.h>
#include <hip/hip_bf16.h>

// Problem constants (from reference): K=5, STRIDE=2, PAD=2
#define EPSV   1e-20f
#define HH     192
#define WWID   192
#define HOv    96
#define WOv    96
#define CINv   32
#define COUTv  32
#define BB     4
#define PLANE_IN   (HH * WWID)     // 36864
#define PLANE_OUT  (HOv * WOv)     // 9216
#define NOUT       (BB * COUTv * PLANE_OUT)  // 1179648 per output tensor
#define TILES_PER_BATCH (PLANE_OUT / 16)     // 576

typedef float v2f __attribute__((ext_vector_type(2)));
typedef float v8f __attribute__((ext_vector_type(8)));

__device__ __forceinline__ float softplusf(float x) {
    // numerically stable softplus
    return fmaxf(x, 0.f) + log1pf(expf(-fabsf(x)));
}

// ---------------------------------------------------------------------------
// Kernel 0: global softplus-sums of spatial_weight (32*25) and channel_weight
// (32*32) -> wsbuf[0], wsbuf[1]
// ---------------------------------------------------------------------------
__global__ __launch_bounds__(256) void reduce_weights(
    const float* __restrict__ sw_raw, const float* __restrict__ cw_raw,
    float* __restrict__ wsbuf) {
    __shared__ float red[256];
    const int t = threadIdx.x;

    float a = 0.f;
    for (int i = t; i < CINv * 25; i += 256) a += softplusf(sw_raw[i]);
    red[t] = a;
    __syncthreads();
    for (int s = 128; s > 0; s >>= 1) {
        if (t < s) red[t] += red[t + s];
        __syncthreads();
    }
    if (t == 0) wsbuf[0] = red[0];
    __syncthreads();

    float b = 0.f;
    for (int i = t; i < COUTv * CINv; i += 256) b += softplusf(cw_raw[i]);
    red[t] = b;
    __syncthreads();
    for (int s = 128; s > 0; s >>= 1) {
        if (t < s) red[t] += red[t + s];
        __syncthreads();
    }
    if (t == 0) wsbuf[1] = red[0];
}

// ---------------------------------------------------------------------------
// Kernel 1: fused 5x5/stride-2 windowed argmax-gather + depthwise conv.
// One thread per (b, c, ho, wo). Block covers 256 consecutive output pixels
// of one (b,c) plane; softplus(spatial_weight[c]) staged in LDS.
// The cd/(d+eps) argmax is done divide-free via cross-multiplication
// (both denominators strictly positive), preserving first-max semantics.
// Writes planar [B][C][9216]:
//   X1 = cs_spatial * s_spatial, X2 = cs_spatial, SFD = s_from_d, CSFD = cs_from_d
// ---------------------------------------------------------------------------
__global__ __launch_bounds__(256) void spatial_kernel(
    const float* __restrict__ d,  const float* __restrict__ cd,
    const float* __restrict__ s,  const float* __restrict__ cs,
    const float* __restrict__ w_s_from_d,
    const float* __restrict__ spatial_weight,
    const float* __restrict__ wsbuf,
    float* __restrict__ X1, float* __restrict__ X2,
    float* __restrict__ SFD, float* __restrict__ CSFD) {

    const int plane = blockIdx.x / 36;                       // b*32 + c
    const int p     = (blockIdx.x % 36) * 256 + threadIdx.x; // output pixel
    const int c     = plane & (CINv - 1);

    __shared__ float swS[25];
    if (threadIdx.x < 25)
        swS[threadIdx.x] = softplusf(spatial_weight[c * 25 + threadIdx.x]);
    __syncthreads();

    const int ho = p / WOv;
    const int wo = p % WOv;

    const float* __restrict__ dp  = d  + (size_t)plane * PLANE_IN;
    const float* __restrict__ cdp = cd + (size_t)plane * PLANE_IN;
    const float* __restrict__ sp  = s  + (size_t)plane * PLANE_IN;
    const float* __restrict__ csp = cs + (size_t)plane * PLANE_IN;

    const int hbase = ho * 2 - 2;
    const int wbase = wo * 2 - 2;

    float v1 = -3.4e38f;                  // max of d*cd
    float d1 = 0.f, cd1 = 0.f;            // d,cd at j_max
    float d2 = 0.f, cd2 = -1.f;           // d,cd at argmax cd/(d+eps) (cd2<0 => first tap wins)
    float nom = 0.f, den = 0.f;           // depthwise conv accumulators

    // Hint next rows into cache (gfx1250 global_prefetch_b8)
    if (hbase + 5 >= 0 && hbase + 5 < HH)
        __builtin_prefetch(dp + (hbase + 5) * WWID + (wbase < 0 ? 0 : wbase), 0, 0);

#pragma unroll
    for (int ky = 0; ky < 5; ++ky) {
        const int hi = hbase + ky;
        if ((unsigned)hi < (unsigned)HH) {
#pragma unroll
            for (int kx = 0; kx < 5; ++kx) {
                const int wi = wbase + kx;
                if ((unsigned)wi < (unsigned)WWID) {
                    const int idx = hi * WWID + wi;
                    const float dv  = dp[idx];
                    const float cdv = cdp[idx];
                    const float sv  = sp[idx];
                    const float csv = csp[idx];
                    const float a1 = dv * cdv;
                    if (a1 > v1) { v1 = a1; d1 = dv; cd1 = cdv; }
                    // cdv/(dv+eps) > cd2/(d2+eps)  <=>  cdv*(d2+eps) > cd2*(dv+eps)
                    if (cdv * (d2 + EPSV) > cd2 * (dv + EPSV)) { d2 = dv; cd2 = cdv; }
                    const float w = swS[ky * 5 + kx];
                    nom = fmaf(w, csv * sv, nom);
                    den = fmaf(w, csv, den);
                }
            }
        }
    }

    const float wsfd = 1.f / (1.f + expf(-w_s_from_d[0]));  // sigmoid
    const float mdm  = fabsf(d2 / (d1 + EPSV));             // |d_min / d_max|
    const float sfd  = (1.f - wsfd) * mdm + wsfd * mdm * mdm;
    const float csfd = cd1 * cd2;                           // cd_max * cd_min

    const float ssp  = nom / (den + EPSV);                  // s_spatial
    const float cssp = den / (wsbuf[0] + EPSV);             // cs_spatial

    const size_t gi = (size_t)plane * PLANE_OUT + p;
    X1[gi]   = cssp * ssp;
    X2[gi]   = cssp;
    SFD[gi]  = sfd;
    CSFD[gi] = csfd;
}

// ---------------------------------------------------------------------------
// Kernel 2: per-pixel 32x32 channel mix as fp32 WMMA GEMM + fused epilogue.
// One wave handles 16 pixels: D[cout,pix] = softplus(cw)[cout,cin] x X[cin,pix]
// for both X1 (nom) and X2 (den), cout split into two 16-row halves.
// VGPR layouts per ISA 7.12.2 (f32 16x4 A / 4x16 B / 16x16 C).
// softplus(cw), softplus(w_prop) and bias are staged in LDS once per block
// so the unrolled epilogue does no transcendentals except one sigmoid/element.
// ---------------------------------------------------------------------------
__global__ __launch_bounds__(128) void chanmix_kernel(
    const float* __restrict__ X1, const float* __restrict__ X2,
    const float* __restrict__ SFD, const float* __restrict__ CSFD,
    const float* __restrict__ channel_weight,
    const float* __restrict__ w_prop, const float* __restrict__ bias,
    const float* __restrict__ wsbuf,
    float* __restrict__ out_s, float* __restrict__ out_cs) {

    __shared__ float cwS[COUTv * CINv];
    __shared__ float wpS[COUTv];
    __shared__ float biasS[COUTv];
    for (int i = threadIdx.x; i < COUTv * CINv; i += 128)
        cwS[i] = softplusf(channel_weight[i]);
    if (threadIdx.x < COUTv) {
        wpS[threadIdx.x]   = softplusf(w_prop[threadIdx.x]);
        biasS[threadIdx.x] = bias[threadIdx.x];
    }
    __syncthreads();

    const int wave   = threadIdx.x >> 5;
    const int lane   = threadIdx.x & 31;
    const int tile   = blockIdx.x * 4 + wave;        // 0..2303
    const int bb     = tile / TILES_PER_BATCH;       // batch
    const int pbase  = (tile % TILES_PER_BATCH) * 16;
    const int n      = lane & 15;                    // pixel within tile (N)
    const int hiHalf = lane >> 4;                    // lane half selects K pair
    const int p      = pbase + n;

    const float* __restrict__ x1 = X1 + (size_t)bb * CINv * PLANE_OUT;
    const float* __restrict__ x2 = X2 + (size_t)bb * CINv * PLANE_OUT;

    v8f acc00 = {};  // cout 0..15,  nom
    v8f acc01 = {};  // cout 0..15,  den
    v8f acc10 = {};  // cout 16..31, nom
    v8f acc11 = {};  // cout 16..31, den

    const int m = lane & 15;  // A-matrix row (within 16-cout half)

#pragma unroll
    for (int k0 = 0; k0 < CINv; k0 += 4) {
        const int kb = k0 + (hiHalf << 1);  // K base for this lane half
        v2f b1, b2, a0, a1;
        // B operand: X[k, pixel] — consecutive lanes hit consecutive pixels
        b1.x = x1[kb * PLANE_OUT + p];
        b1.y = x1[(kb + 1) * PLANE_OUT + p];
        b2.x = x2[kb * PLANE_OUT + p];
        b2.y = x2[(kb + 1) * PLANE_OUT + p];
        // A operand: softplus(cw)[cout, k] from LDS
        a0.x = cwS[m * CINv + kb];
        a0.y = cwS[m * CINv + kb + 1];
        a1.x = cwS[(16 + m) * CINv + kb];
        a1.y = cwS[(16 + m) * CINv + kb + 1];

        acc00 = __builtin_amdgcn_wmma_f32_16x16x4_f32(false, a0, false, b1,
                                                      (short)0, acc00, false, false);
        acc01 = __builtin_amdgcn_wmma_f32_16x16x4_f32(false, a0, false, b2,
                                                      (short)0, acc01, false, false);
        acc10 = __builtin_amdgcn_wmma_f32_16x16x4_f32(false, a1, false, b1,
                                                      (short)0, acc10, false, false);
        acc11 = __builtin_amdgcn_wmma_f32_16x16x4_f32(false, a1, false, b2,
                                                      (short)0, acc11, false, false);
    }

    const float sumcw = wsbuf[1];

    // C/D layout: VGPR r, lanes 0-15 -> M=r; lanes 16-31 -> M=r+8; N=lane&15.
#pragma unroll
    for (int h = 0; h < 2; ++h) {
        const v8f an = h ? acc10 : acc00;
        const v8f ad = h ? acc11 : acc01;
#pragma unroll
        for (int r = 0; r < 8; ++r) {
            const int o    = h * 16 + hiHalf * 8 + r;   // output channel
            const float nomv = an[r];
            const float denv = ad[r];
            const float so  = nomv / (denv + EPSV);
            const float cso = denv / (sumcw + EPSV);

            const size_t gi = ((size_t)bb * COUTv + o) * PLANE_OUT + p;
            const float sfd  = SFD[gi];
            const float csfd = CSFD[gi];
            const float wp   = wpS[o];

            float s2  = (wp * cso * so + csfd * sfd) / (wp * cso + csfd + EPSV);
            float cs2 = (wp * cso + csfd) / (wp + 1.f);
            s2 += biasS[o];
            s2 = 0.5f / (1.f + expf(-(2.f * s2 - 1.f))) + 0.5f;  // sigmoid(2x-1)/2+0.5

            out_s[gi]  = s2;
            out_cs[gi] = cs2 * 0.25f;  // / (STRIDE*STRIDE)
        }
    }
}

// ---------------------------------------------------------------------------
// Host-side launcher
// ---------------------------------------------------------------------------
extern "C" void kernel_launch(void* const* d_in, const int* in_sizes, int n_in,
                              void* d_out, int out_size, void* d_ws, size_t ws_size,
                              hipStream_t stream) {
    const float* d   = (const float*)d_in[0];
    const float* cd  = (const float*)d_in[1];
    const float* s   = (const float*)d_in[2];
    const float* cs  = (const float*)d_in[3];
    const float* w_s_from_d    = (const float*)d_in[4];
    const float* w_prop        = (const float*)d_in[5];
    const float* spatial_w     = (const float*)d_in[6];
    const float* channel_w     = (const float*)d_in[7];
    const float* bias          = (const float*)d_in[8];

    float* ws    = (float*)d_ws;
    float* wsbuf = ws;            // [0]=sum softplus(sw), [1]=sum softplus(cw)
    float* X1    = ws + 16;
    float* X2    = X1 + NOUT;
    float* SFD   = X2 + NOUT;
    float* CSFD  = SFD + NOUT;

    float* out_s  = (float*)d_out;
    float* out_cs = out_s + NOUT;

    reduce_weights<<<1, 256, 0, stream>>>(spatial_w, channel_w, wsbuf);

    spatial_kernel<<<BB * CINv * 36, 256, 0, stream>>>(
        d, cd, s, cs, w_s_from_d, spatial_w, wsbuf, X1, X2, SFD, CSFD);

    chanmix_kernel<<<TILES_PER_BATCH * BB / 4, 128, 0, stream>>>(
        X1, X2, SFD, CSFD, channel_w, w_prop, bias, wsbuf, out_s, out_cs);
}